// PointNetOrig2Encoder_66520453480976
// MI455X (gfx1250) — compile-verified
//
#include <hip/hip_runtime.h>
#include <hip/hip_bf16.h>
#include <stdint.h>

// ---------------------------------------------------------------------------
// PointNet++ (SSG) encoder for gfx1250.
// All GEMMs use V_WMMA_F32_16X16X4_F32 (exact fp32 path on CDNA5).
// ---------------------------------------------------------------------------

typedef __attribute__((ext_vector_type(2))) float v2f;
typedef __attribute__((ext_vector_type(8))) float v8f;

__device__ __forceinline__ v8f wmma_f32(v2f a, v2f b, v8f c) {
  // 8 args: (neg_a, A, neg_b, B, c_mod, C, reuse_a, reuse_b)
  return __builtin_amdgcn_wmma_f32_16x16x4_f32(
      false, a, false, b, (short)0, c, false, false);
}

#define EPS_BN 1e-5f

// ===========================================================================
// Farthest point sampling: one block per batch. Points + running min-dist in
// LDS; 256-thread argmax tree reduction each iteration (ties -> lowest index,
// matching jnp.argmax). Also gathers new_xyz on the fly.
// ===========================================================================
__global__ __launch_bounds__(256) void fps_kernel(
    const float* __restrict__ xyz, int N, int npoint,
    int* __restrict__ fidx, float* __restrict__ new_xyz) {
  __shared__ float px[2048], py[2048], pz[2048], dist[2048];
  __shared__ float rbd[256];
  __shared__ int rbi[256];

  const int b = blockIdx.x, tid = threadIdx.x;
  const float* P = xyz + (size_t)b * N * 3;

  for (int i = tid; i < N; i += 256) {
    px[i] = P[i * 3 + 0];
    py[i] = P[i * 3 + 1];
    pz[i] = P[i * 3 + 2];
    dist[i] = 1e10f;
  }
  __syncthreads();

  int far = 0;
  for (int it = 0; it < npoint; ++it) {
    if (tid == 0) {
      fidx[b * npoint + it] = far;
      new_xyz[((size_t)b * npoint + it) * 3 + 0] = px[far];
      new_xyz[((size_t)b * npoint + it) * 3 + 1] = py[far];
      new_xyz[((size_t)b * npoint + it) * 3 + 2] = pz[far];
    }
    const float fx = px[far], fy = py[far], fz = pz[far];
    float bd = -1.0f;
    int bi = 0;
    for (int i = tid; i < N; i += 256) {
      const float dx = px[i] - fx, dy = py[i] - fy, dz = pz[i] - fz;
      float dm = fminf(dist[i], dx * dx + dy * dy + dz * dz);
      dist[i] = dm;
      if (dm > bd) { bd = dm; bi = i; }
    }
    rbd[tid] = bd;
    rbi[tid] = bi;
    __syncthreads();
    for (int off = 128; off > 0; off >>= 1) {
      if (tid < off) {
        const float od = rbd[tid + off];
        const int oi = rbi[tid + off];
        if (od > rbd[tid] || (od == rbd[tid] && oi < rbi[tid])) {
          rbd[tid] = od;
          rbi[tid] = oi;
        }
      }
      __syncthreads();
    }
    far = rbi[0];
    __syncthreads();
  }
}

// ===========================================================================
// Ball query: one wave per query point. Scans N points in lane-sized chunks,
// compacting in-range indices in ascending order via ballot/popc (= the
// reference's sort-then-take-first-nsample), padding with the first hit.
// ===========================================================================
__global__ __launch_bounds__(256) void ball_query_kernel(
    const float* __restrict__ xyz, const float* __restrict__ new_xyz,
    int* __restrict__ gidx, int B, int N, int S, int nsample, float r2) {
  const int lane = threadIdx.x & 31;
  const int wave = threadIdx.x >> 5;
  const int q = blockIdx.x * 8 + wave;
  if (q >= B * S) return;
  const int b = q / S;
  const float qx = new_xyz[(size_t)q * 3 + 0];
  const float qy = new_xyz[(size_t)q * 3 + 1];
  const float qz = new_xyz[(size_t)q * 3 + 2];
  const float* P = xyz + (size_t)b * N * 3;
  int* out = gidx + (size_t)q * nsample;

  int cnt = 0;
  int firstIdx = 0;
  bool haveFirst = false;
  for (int base = 0; base < N && cnt < nsample; base += 32) {
    const int p = base + lane;
    bool in = false;
    if (p < N) {
      const float dx = P[p * 3 + 0] - qx;
      const float dy = P[p * 3 + 1] - qy;
      const float dz = P[p * 3 + 2] - qz;
      in = (dx * dx + dy * dy + dz * dz) <= r2;
    }
    const unsigned mask = (unsigned)__ballot(in);
    if (!haveFirst && mask) {
      firstIdx = base + __builtin_ctz(mask);
      haveFirst = true;
    }
    if (in) {
      const int pos = cnt + __builtin_popcount(mask & ((1u << lane) - 1u));
      if (pos < nsample) out[pos] = p;
    }
    cnt += __builtin_popcount(mask);
  }
  if (cnt < nsample) {
    for (int j = cnt + lane; j < nsample; j += 32) out[j] = firstIdx;
  }
}

// ===========================================================================
// WMMA GEMM over LDS activations: Xout = relu((Xin * W^T + b) * g' + beta)
// Xin: LDS [TR][ldin] (ldin padded to mult of 4), W: global [Cout][CinAct].
// ===========================================================================
template <int TR>
__device__ __forceinline__ void lds_gemm(
    const float* __restrict__ Xin, int ldin, int CinAct,
    float* __restrict__ Xout, int Cout,
    const float* __restrict__ W, const float* __restrict__ bias,
    const float* __restrict__ g, const float* __restrict__ be,
    int lane, int wave, int nwaves) {
  const int mt = TR / 16;
  const int ntiles = mt * (Cout / 16);
  const int half = lane >> 4;
  const int l16 = lane & 15;
  const float bnc = rsqrtf(1.0f + EPS_BN);

  for (int t = wave; t < ntiles; t += nwaves) {
    const int m0 = (t % mt) * 16;
    const int n0 = (t / mt) * 16;
    v8f acc = {0.f, 0.f, 0.f, 0.f, 0.f, 0.f, 0.f, 0.f};
    const int arow = (m0 + l16) * ldin;
    const int n = n0 + l16;
    for (int k0 = 0; k0 < ldin; k0 += 4) {
      const int kk = k0 + 2 * half;
      v2f a;
      a.x = Xin[arow + kk];
      a.y = Xin[arow + kk + 1];
      v2f bm;
      bm.x = (kk < CinAct) ? W[(size_t)n * CinAct + kk] : 0.0f;
      bm.y = (kk + 1 < CinAct) ? W[(size_t)n * CinAct + kk + 1] : 0.0f;
      acc = wmma_f32(a, bm, acc);
    }
    const float gb = g[n] * bnc;
    const float bt = be[n];
    const float bi = bias[n];
#pragma unroll
    for (int j = 0; j < 8; ++j) {
      const int row = m0 + j + half * 8;
      float v = (acc[j] + bi) * gb + bt;
      v = fmaxf(v, 0.0f);
      Xout[row * Cout + n] = v;
    }
  }
}

// Same, but the epilogue max-pools rows directly into global memory
// (atomicMax on float bits; everything is post-ReLU >= 0, pool pre-zeroed).
template <int TR>
__device__ __forceinline__ void lds_gemm_pool(
    const float* __restrict__ Xin, int ldin, int CinAct,
    float* __restrict__ pool, int Cout,
    const float* __restrict__ W, const float* __restrict__ bias,
    const float* __restrict__ g, const float* __restrict__ be,
    int lane, int wave, int nwaves) {
  const int mt = TR / 16;
  const int ntiles = mt * (Cout / 16);
  const int half = lane >> 4;
  const int l16 = lane & 15;
  const float bnc = rsqrtf(1.0f + EPS_BN);

  for (int t = wave; t < ntiles; t += nwaves) {
    const int m0 = (t % mt) * 16;
    const int n0 = (t / mt) * 16;
    v8f acc = {0.f, 0.f, 0.f, 0.f, 0.f, 0.f, 0.f, 0.f};
    const int arow = (m0 + l16) * ldin;
    const int n = n0 + l16;
    for (int k0 = 0; k0 < ldin; k0 += 4) {
      const int kk = k0 + 2 * half;
      v2f a;
      a.x = Xin[arow + kk];
      a.y = Xin[arow + kk + 1];
      v2f bm;
      bm.x = (kk < CinAct) ? W[(size_t)n * CinAct + kk] : 0.0f;
      bm.y = (kk + 1 < CinAct) ? W[(size_t)n * CinAct + kk + 1] : 0.0f;
      acc = wmma_f32(a, bm, acc);
    }
    const float gb = g[n] * bnc;
    const float bt = be[n];
    const float bi = bias[n];
    float m = 0.0f;
#pragma unroll
    for (int j = 0; j < 8; ++j) {
      float v = (acc[j] + bi) * gb + bt;
      v = fmaxf(v, 0.0f);
      m = fmaxf(m, v);
    }
    atomicMax((int*)(pool + n), __float_as_int(m));
  }
}

// ===========================================================================
// Fused set-abstraction MLP: build grouped/centred features in LDS, run the
// three MLP layers with WMMA (ping-pong LDS), max-pool layer-3 into pool_out.
// MODE: 1 = SA1 (xyz only), 2 = SA2 (xyz + gathered features), 3 = SA3
// (group_all: identity gather, no centring). TR divides KG, so a block never
// straddles a pooling group.
// ===========================================================================
template <int MODE, int TR, int C0P, int C1, int C2, int C3, int KG>
__global__ __launch_bounds__(256) void sa_kernel(
    const float* __restrict__ src_xyz, const float* __restrict__ src_pts,
    const int* __restrict__ gidx, const float* __restrict__ new_xyz,
    const float* __restrict__ W1, const float* __restrict__ b1,
    const float* __restrict__ g1, const float* __restrict__ be1,
    const float* __restrict__ W2, const float* __restrict__ b2,
    const float* __restrict__ g2, const float* __restrict__ be2,
    const float* __restrict__ W3, const float* __restrict__ b3,
    const float* __restrict__ g3, const float* __restrict__ be3,
    float* __restrict__ pool_out, int S, int Nsrc, int Cpts, int C0A) {
  constexpr int BA = (C0P > C2 ? C0P : C2) * TR;  // holds X0, then X2
  constexpr int BB = C1 * TR;                     // holds X1
  __shared__ float bufA[BA];
  __shared__ float bufB[BB];

  const int tid = threadIdx.x;
  const int row0 = blockIdx.x * TR;

  // ---- build X0 (padded channels zero) ----
  for (int idx = tid; idx < TR * C0P; idx += 256) {
    const int r = idx / C0P;
    const int c = idx - r * C0P;
    const int gr = row0 + r;
    const int b = gr / (S * KG);
    const int rem = gr - b * (S * KG);
    const int s = rem / KG;
    const int k = rem - s * KG;
    float v = 0.0f;
    if (MODE == 3) {
      if (c < 3)
        v = src_xyz[((size_t)b * Nsrc + k) * 3 + c];
      else if (c < 3 + Cpts)
        v = src_pts[((size_t)b * Nsrc + k) * Cpts + (c - 3)];
    } else {
      const int gpt = gidx[((size_t)b * S + s) * KG + k];
      if (c < 3) {
        v = src_xyz[((size_t)b * Nsrc + gpt) * 3 + c] -
            new_xyz[((size_t)b * S + s) * 3 + c];
      } else if (MODE == 2 && c < 3 + Cpts) {
        v = src_pts[((size_t)b * Nsrc + gpt) * Cpts + (c - 3)];
      }
    }
    bufA[r * C0P + c] = v;
  }
  __syncthreads();

  const int lane = tid & 31;
  const int wave = tid >> 5;

  lds_gemm<TR>(bufA, C0P, C0A, bufB, C1, W1, b1, g1, be1, lane, wave, 8);
  __syncthreads();
  lds_gemm<TR>(bufB, C1, C1, bufA, C2, W2, b2, g2, be2, lane, wave, 8);
  __syncthreads();
  const int group = row0 / KG;
  lds_gemm_pool<TR>(bufA, C2, C2, pool_out + (size_t)group * C3, C3, W3, b3,
                    g3, be3, lane, wave, 8);
}

// ===========================================================================
// FC head GEMM: Y = act((X * W^T + b) * g' + beta), M=32 rows, 2 waves/block,
// one 16x16 WMMA tile per wave per block, N tiled over grid.x.
// ===========================================================================
__global__ __launch_bounds__(64) void fc_kernel(
    const float* __restrict__ X, const float* __restrict__ W,
    const float* __restrict__ bias, const float* __restrict__ g,
    const float* __restrict__ be, float* __restrict__ Y, int M, int N, int K,
    int do_relu) {
  const int lane = threadIdx.x & 31;
  const int wave = threadIdx.x >> 5;
  const int m0 = wave * 16;
  if (m0 >= M) return;  // wave-uniform
  const int n0 = blockIdx.x * 16;
  const int half = lane >> 4;
  const int l16 = lane & 15;

  v8f acc = {0.f, 0.f, 0.f, 0.f, 0.f, 0.f, 0.f, 0.f};
  const float* Xr = X + (size_t)(m0 + l16) * K;
  const int n = n0 + l16;
  const float* Wr = W + (size_t)n * K;
  for (int k0 = 0; k0 < K; k0 += 4) {
    const int kk = k0 + 2 * half;
    v2f a;
    a.x = Xr[kk];
    a.y = Xr[kk + 1];
    v2f bm;
    bm.x = Wr[kk];
    bm.y = Wr[kk + 1];
    acc = wmma_f32(a, bm, acc);
  }
  const float bnc = rsqrtf(1.0f + EPS_BN);
  const float gb = g ? g[n] * bnc : 1.0f;
  const float bt = be ? be[n] : 0.0f;
  const float bi = bias[n];
#pragma unroll
  for (int j = 0; j < 8; ++j) {
    const int row = m0 + j + half * 8;
    float v = (acc[j] + bi) * gb + bt;
    if (do_relu) v = fmaxf(v, 0.0f);
    Y[(size_t)row * N + n] = v;
  }
}

// ===========================================================================
// Host orchestration
// ===========================================================================
extern "C" void kernel_launch(void* const* d_in, const int* in_sizes, int n_in,
                              void* d_out, int out_size, void* d_ws,
                              size_t ws_size, hipStream_t stream) {
  (void)in_sizes; (void)n_in; (void)out_size; (void)ws_size;

  constexpr int B = 32, N1 = 2048;
  constexpr int NP1 = 512, NS1 = 32;
  constexpr int NP2 = 128, NS2 = 64;

  const float* xyz = (const float*)d_in[0];
  const float* sa1_w[3] = {(const float*)d_in[1], (const float*)d_in[2], (const float*)d_in[3]};
  const float* sa1_b[3] = {(const float*)d_in[4], (const float*)d_in[5], (const float*)d_in[6]};
  const float* sa1_g[3] = {(const float*)d_in[7], (const float*)d_in[8], (const float*)d_in[9]};
  const float* sa1_e[3] = {(const float*)d_in[10], (const float*)d_in[11], (const float*)d_in[12]};
  const float* sa2_w[3] = {(const float*)d_in[13], (const float*)d_in[14], (const float*)d_in[15]};
  const float* sa2_b[3] = {(const float*)d_in[16], (const float*)d_in[17], (const float*)d_in[18]};
  const float* sa2_g[3] = {(const float*)d_in[19], (const float*)d_in[20], (const float*)d_in[21]};
  const float* sa2_e[3] = {(const float*)d_in[22], (const float*)d_in[23], (const float*)d_in[24]};
  const float* sa3_w[3] = {(const float*)d_in[25], (const float*)d_in[26], (const float*)d_in[27]};
  const float* sa3_b[3] = {(const float*)d_in[28], (const float*)d_in[29], (const float*)d_in[30]};
  const float* sa3_g[3] = {(const float*)d_in[31], (const float*)d_in[32], (const float*)d_in[33]};
  const float* sa3_e[3] = {(const float*)d_in[34], (const float*)d_in[35], (const float*)d_in[36]};
  const float* fc1_w = (const float*)d_in[37];
  const float* fc1_b = (const float*)d_in[38];
  const float* bn1_g = (const float*)d_in[39];
  const float* bn1_b = (const float*)d_in[40];
  const float* fc2_w = (const float*)d_in[41];
  const float* fc2_b = (const float*)d_in[42];
  const float* bn2_g = (const float*)d_in[43];
  const float* bn2_b = (const float*)d_in[44];
  const float* fc3_w = (const float*)d_in[45];
  const float* fc3_b = (const float*)d_in[46];

  // --- workspace carve-out ---
  uint8_t* wsp = (uint8_t*)d_ws;
  size_t off = 0;
  auto alloc = [&](size_t bytes) -> void* {
    off = (off + 255) & ~(size_t)255;
    void* p = wsp + off;
    off += bytes;
    return p;
  };
  int* fidx1 = (int*)alloc((size_t)B * NP1 * 4);
  float* l1_xyz = (float*)alloc((size_t)B * NP1 * 3 * 4);
  int* gidx1 = (int*)alloc((size_t)B * NP1 * NS1 * 4);
  float* l1_pts = (float*)alloc((size_t)B * NP1 * 128 * 4);
  int* fidx2 = (int*)alloc((size_t)B * NP2 * 4);
  float* l2_xyz = (float*)alloc((size_t)B * NP2 * 3 * 4);
  int* gidx2 = (int*)alloc((size_t)B * NP2 * NS2 * 4);
  float* l2_pts = (float*)alloc((size_t)B * NP2 * 256 * 4);
  float* l3 = (float*)alloc((size_t)B * 1024 * 4);
  float* fc1o = (float*)alloc((size_t)B * 512 * 4);
  float* fc2o = (float*)alloc((size_t)B * 256 * 4);

  // ---- SA1 ----
  fps_kernel<<<B, 256, 0, stream>>>(xyz, N1, NP1, fidx1, l1_xyz);
  ball_query_kernel<<<(B * NP1) / 8, 256, 0, stream>>>(
      xyz, l1_xyz, gidx1, B, N1, NP1, NS1, 0.2f * 0.2f);
  hipMemsetAsync(l1_pts, 0, (size_t)B * NP1 * 128 * 4, stream);
  // rows = B*NP1*NS1 = 524288, TR=32 -> 16384 blocks
  sa_kernel<1, 32, 4, 64, 64, 128, NS1><<<(B * NP1 * NS1) / 32, 256, 0, stream>>>(
      xyz, nullptr, gidx1, l1_xyz,
      sa1_w[0], sa1_b[0], sa1_g[0], sa1_e[0],
      sa1_w[1], sa1_b[1], sa1_g[1], sa1_e[1],
      sa1_w[2], sa1_b[2], sa1_g[2], sa1_e[2],
      l1_pts, NP1, N1, 0, 3);

  // ---- SA2 ----
  fps_kernel<<<B, 256, 0, stream>>>(l1_xyz, NP1, NP2, fidx2, l2_xyz);
  ball_query_kernel<<<(B * NP2) / 8, 256, 0, stream>>>(
      l1_xyz, l2_xyz, gidx2, B, NP1, NP2, NS2, 0.4f * 0.4f);
  hipMemsetAsync(l2_pts, 0, (size_t)B * NP2 * 256 * 4, stream);
  // rows = B*NP2*NS2 = 262144, TR=32 -> 8192 blocks
  sa_kernel<2, 32, 132, 128, 128, 256, NS2><<<(B * NP2 * NS2) / 32, 256, 0, stream>>>(
      l1_xyz, l1_pts, gidx2, l2_xyz,
      sa2_w[0], sa2_b[0], sa2_g[0], sa2_e[0],
      sa2_w[1], sa2_b[1], sa2_g[1], sa2_e[1],
      sa2_w[2], sa2_b[2], sa2_g[2], sa2_e[2],
      l2_pts, NP2, NP1, 128, 131);

  // ---- SA3 (group_all) ----
  hipMemsetAsync(l3, 0, (size_t)B * 1024 * 4, stream);
  // rows = B*128 = 4096, TR=16 -> 256 blocks
  sa_kernel<3, 16, 260, 256, 512, 1024, NP2><<<(B * NP2) / 16, 256, 0, stream>>>(
      l2_xyz, l2_pts, nullptr, nullptr,
      sa3_w[0], sa3_b[0], sa3_g[0], sa3_e[0],
      sa3_w[1], sa3_b[1], sa3_g[1], sa3_e[1],
      sa3_w[2], sa3_b[2], sa3_g[2], sa3_e[2],
      l3, 1, NP2, 256, 259);

  // ---- FC head ----
  fc_kernel<<<512 / 16, 64, 0, stream>>>(l3, fc1_w, fc1_b, bn1_g, bn1_b, fc1o,
                                         B, 512, 1024, 1);
  fc_kernel<<<256 / 16, 64, 0, stream>>>(fc1o, fc2_w, fc2_b, bn2_g, bn2_b,
                                         fc2o, B, 256, 512, 1);
  fc_kernel<<<256 / 16, 64, 0, stream>>>(fc2o, fc3_w, fc3_b, nullptr, nullptr,
                                         (float*)d_out, B, 256, 256, 0);
}